// SelfAttentionHybrid_65481071400803
// MI455X (gfx1250) — compile-verified
//
#include <hip/hip_runtime.h>
#include <hip/hip_bf16.h>

typedef __attribute__((ext_vector_type(2))) float v2f;
typedef __attribute__((ext_vector_type(8))) float v8f;

#define BATCH 8192
#define BN_EPS 1e-5f

// ---------------- conv1: x[B,1,28,28] -> conv3x3 SAME -> relu -> maxpool2 -> f1[B,8,14,14]
__global__ void conv1_kernel(const float* __restrict__ x,
                             const float* __restrict__ w,     // [8,1,3,3]
                             const float* __restrict__ bias,  // [8]
                             float* __restrict__ f1) {
    int idx = blockIdx.x * blockDim.x + threadIdx.x;
    const int total = BATCH * 8 * 14 * 14;
    if (idx >= total) return;
    int pw = idx % 14;
    int ph = (idx / 14) % 14;
    int c  = (idx / 196) % 8;
    int b  = idx / 1568;

    float wk[9];
#pragma unroll
    for (int i = 0; i < 9; ++i) wk[i] = w[c * 9 + i];
    float bc = bias[c];
    const float* xb = x + (size_t)b * 784;

    float best = 0.0f;  // relu outputs are >= 0
#pragma unroll
    for (int dy = 0; dy < 2; ++dy) {
#pragma unroll
        for (int dx = 0; dx < 2; ++dx) {
            int h  = 2 * ph + dy;
            int wc = 2 * pw + dx;
            float acc = bc;
#pragma unroll
            for (int ky = 0; ky < 3; ++ky) {
                int yy = h + ky - 1;
                if (yy < 0 || yy >= 28) continue;
#pragma unroll
                for (int kx = 0; kx < 3; ++kx) {
                    int xx = wc + kx - 1;
                    if (xx < 0 || xx >= 28) continue;
                    acc = fmaf(xb[yy * 28 + xx], wk[ky * 3 + kx], acc);
                }
            }
            best = fmaxf(best, fmaxf(acc, 0.0f));
        }
    }
    f1[idx] = best;
}

// ---------------- conv2 as shift-decomposed implicit GEMM on V_WMMA_F32_16X16X4_F32.
// Per image: C[pos,oc] = sum_{ky,kx} sum_ic in_pad[y+ky][x+kx][ic] * w[oc][ic][ky][kx]
//   M = 196 positions (13 tiles of 16, padded to 208), N = 16 oc (exact), K = 8 ic per
//   shift (2 WMMA K=4 steps) x 9 shifts = 18 chained WMMAs per M-tile.
// Then relu + 2x2 maxpool -> f2[B,784] (flatten order oc*49 + ph*7 + pw).
__global__ void conv2_wmma_kernel(const float* __restrict__ f1,  // [B,8,14,14]
                                  const float* __restrict__ w,   // [16,8,3,3]
                                  const float* __restrict__ bias,// [16]
                                  float* __restrict__ f2) {      // [B,784]
    __shared__ float in_pad[18 * 16 * 8];   // [y 0..17][x 0..15][ic 0..7], zero-padded halo
    __shared__ float wB2[9 * 16 * 10];      // [shift][oc][ic padded to 10] (conflict-free b64)
    __shared__ float conv_out[208 * 20];    // [pos 0..207][oc padded to 20]; rows >=196 are
                                            // scratch for the padded tile (never read back)

    const int b = blockIdx.x;
    const int t = threadIdx.x;              // 128 threads = 4 wave32
    const int lane = t & 31;
    const int wave = t >> 5;

    // Stage zero-padded input tile: in_pad[y][x][ic]
    const float* src = f1 + (size_t)b * 1568;
    for (int i = t; i < 18 * 16 * 8; i += 128) {
        int ic = i & 7;
        int x  = (i >> 3) & 15;
        int y  = i >> 7;
        float v = 0.0f;
        if (y >= 1 && y <= 14 && x >= 1 && x <= 14)
            v = src[ic * 196 + (y - 1) * 14 + (x - 1)];
        in_pad[i] = v;
    }
    // Stage transposed weights: wB2[shift][oc][ic]
    for (int i = t; i < 9 * 16 * 8; i += 128) {
        int ic    = i & 7;
        int oc    = (i >> 3) & 15;
        int shift = i >> 7;               // ky*3+kx
        wB2[shift * 160 + oc * 10 + ic] = w[oc * 72 + ic * 9 + shift];
    }
    __syncthreads();

    const int mn   = lane & 15;           // A-row / B-col index
    const int half = lane >> 4;
    const int koff = half << 1;           // K offset {0,2} within each K=4 step
    const float bc = bias[mn];            // per-lane oc bias (n = lane&15)

    for (int tile = wave; tile < 13; tile += 4) {
        int pos = tile * 16 + mn;         // 0..207 (>=196 computes on zero halo rows)
        int y = pos / 14;
        int x = pos - y * 14;
        v8f acc = {};
#pragma unroll
        for (int ky = 0; ky < 3; ++ky) {
#pragma unroll
            for (int kx = 0; kx < 3; ++kx) {
                const float* apix = in_pad + (((y + ky) << 4) + (x + kx)) * 8 + koff;
                const float* brow = wB2 + (ky * 3 + kx) * 160 + mn * 10 + koff;
#pragma unroll
                for (int s = 0; s < 2; ++s) {   // K = 8 ic -> two K=4 WMMA steps
                    v2f a  = *(const v2f*)(apix + 4 * s);  // ds_load_b64 (pairs -> 2addr)
                    v2f bb = *(const v2f*)(brow + 4 * s);  // register-resident after hoist
                    acc = __builtin_amdgcn_wmma_f32_16x16x4_f32(
                        false, a, false, bb, (short)0, acc, false, false);
                }
            }
        }
        // D layout: VGPR r -> row r (lanes 0-15) / row r+8 (lanes 16-31), col = lane&15.
        // Unconditional spill: rows 196..207 land in conv_out's scratch tail.
#pragma unroll
        for (int r = 0; r < 8; ++r) {
            conv_out[(tile * 16 + r + 8 * half) * 20 + mn] = acc[r] + bc;
        }
    }
    __syncthreads();

    // relu + 2x2 maxpool, write in flatten order oc*49 + ph*7 + pw
    for (int o = t; o < 784; o += 128) {
        int pw = o % 7;
        int ph = (o / 7) % 7;
        int oc = o / 49;
        int p0 = (2 * ph) * 14 + 2 * pw;
        float m0 = fmaxf(conv_out[p0 * 20 + oc],        conv_out[(p0 + 1) * 20 + oc]);
        float m1 = fmaxf(conv_out[(p0 + 14) * 20 + oc], conv_out[(p0 + 15) * 20 + oc]);
        f2[(size_t)b * 784 + o] = fmaxf(fmaxf(m0, m1), 0.0f);
    }
}

// ---------------- qkv GEMM via V_WMMA_F32_16X16X4_F32 (full fp32 matrix core path)
#define WROW 788  // 784 + 4 pad -> LDS bank stride 20 mod 64: conflict-free
__global__ void qkv_wmma_kernel(const float* __restrict__ f2,     // [B,784]
                                const float* __restrict__ qkv_w,  // [12,784]
                                const float* __restrict__ qkv_b,  // [12]
                                float* __restrict__ qkv) {        // [B,16]
    __shared__ float wlds[16 * WROW];
    int t = threadIdx.x;  // 128 threads = 4 wave32
    for (int i = t; i < 16 * 784; i += 128) {
        int n = i / 784;
        int k = i - n * 784;
        wlds[n * WROW + k] = (n < 12) ? qkv_w[i] : 0.0f;
    }
    __syncthreads();

    int lane = t & 31;
    int wave = t >> 5;
    int b0 = blockIdx.x * 64 + wave * 16;

    int mn   = lane & 15;
    int koff = (lane >> 4) << 1;
    const float* arow = f2 + (size_t)(b0 + mn) * 784 + koff;
    const float* brow = wlds + mn * WROW + koff;

    v8f acc = {};
#pragma unroll 4
    for (int k0 = 0; k0 < 784; k0 += 4) {
        v2f a  = *(const v2f*)(arow + k0);   // global_load_b64
        v2f bb = *(const v2f*)(brow + k0);   // ds_load_b64
        acc = __builtin_amdgcn_wmma_f32_16x16x4_f32(
            false, a, false, bb, (short)0, acc, false, false);
    }

    float bn = (mn < 12) ? qkv_b[mn] : 0.0f;
    int mbase = (lane >> 4) * 8;
#pragma unroll
    for (int r = 0; r < 8; ++r) {
        qkv[(size_t)(b0 + mbase + r) * 16 + mn] = acc[r] + bn;
    }
}

// ---------------- attention epilogue (circuit expvals depend only on params; q,k drop out)
__global__ void attn_out_kernel(const float* __restrict__ params,  // [B,4]
                                const float* __restrict__ qkv,     // [B,16]
                                const float* __restrict__ out_w,   // [4,4]
                                const float* __restrict__ out_b,   // [4]
                                float* __restrict__ out_pre) {     // [B,4]
    int b = blockIdx.x * blockDim.x + threadIdx.x;
    if (b >= BATCH) return;
    float l[4];
    float cp = 1.0f;
#pragma unroll
    for (int i = 0; i < 4; ++i) {
        cp *= cosf(params[b * 4 + i]);
        l[i] = cp;
    }
    float m = fmaxf(fmaxf(l[0], l[1]), fmaxf(l[2], l[3]));
    float e[4], s = 0.0f;
#pragma unroll
    for (int i = 0; i < 4; ++i) { e[i] = expf(l[i] - m); s += e[i]; }
    float inv = 1.0f / s;
    float av[4];
#pragma unroll
    for (int i = 0; i < 4; ++i) av[i] = e[i] * inv * qkv[b * 16 + 8 + i];
#pragma unroll
    for (int c = 0; c < 4; ++c) {
        float acc = out_b[c];
#pragma unroll
        for (int w = 0; w < 4; ++w) acc = fmaf(av[w], out_w[c * 4 + w], acc);
        out_pre[b * 4 + c] = acc;
    }
}

// ---------------- BatchNorm stats: single block, fixed-order (deterministic) tree reduction
__global__ void bn_stats_kernel(const float* __restrict__ out_pre,  // [B,4]
                                float* __restrict__ stats) {        // [8]: mu[4], rstd[4]
    __shared__ float red[1024 * 8];
    int t = threadIdx.x;  // 1024 threads
    float s[4] = {0, 0, 0, 0}, ss[4] = {0, 0, 0, 0};
    for (int b = t; b < BATCH; b += 1024) {
#pragma unroll
        for (int c = 0; c < 4; ++c) {
            float v = out_pre[b * 4 + c];
            s[c] += v;
            ss[c] += v * v;
        }
    }
#pragma unroll
    for (int c = 0; c < 4; ++c) { red[t * 8 + c] = s[c]; red[t * 8 + 4 + c] = ss[c]; }
    __syncthreads();
    for (int off = 512; off > 0; off >>= 1) {
        if (t < off) {
#pragma unroll
            for (int j = 0; j < 8; ++j) red[t * 8 + j] += red[(t + off) * 8 + j];
        }
        __syncthreads();
    }
    if (t == 0) {
        const float invB = 1.0f / (float)BATCH;
#pragma unroll
        for (int c = 0; c < 4; ++c) {
            float mu  = red[c] * invB;
            float var = red[4 + c] * invB - mu * mu;
            stats[c] = mu;
            stats[4 + c] = 1.0f / sqrtf(var + BN_EPS);
        }
    }
}

__global__ void bn_apply_kernel(const float* __restrict__ out_pre,
                                const float* __restrict__ stats,
                                const float* __restrict__ gamma,
                                const float* __restrict__ beta,
                                float* __restrict__ out) {
    int i = blockIdx.x * blockDim.x + threadIdx.x;
    if (i >= BATCH * 4) return;
    int c = i & 3;
    out[i] = (out_pre[i] - stats[c]) * stats[4 + c] * gamma[c] + beta[c];
}

extern "C" void kernel_launch(void* const* d_in, const int* in_sizes, int n_in,
                              void* d_out, int out_size, void* d_ws, size_t ws_size,
                              hipStream_t stream) {
    (void)in_sizes; (void)n_in; (void)out_size; (void)ws_size;
    const float* x      = (const float*)d_in[0];
    const float* params = (const float*)d_in[1];
    const float* c1w    = (const float*)d_in[2];
    const float* c1b    = (const float*)d_in[3];
    const float* c2w    = (const float*)d_in[4];
    const float* c2b    = (const float*)d_in[5];
    const float* qw     = (const float*)d_in[6];
    const float* qb     = (const float*)d_in[7];
    const float* ow     = (const float*)d_in[8];
    const float* ob     = (const float*)d_in[9];
    const float* gamma  = (const float*)d_in[10];
    const float* beta   = (const float*)d_in[11];

    char* ws = (char*)d_ws;
    size_t off = 0;
    float* f1   = (float*)(ws + off); off += (size_t)BATCH * 8 * 196 * 4;   // 51,380,224 B
    float* f2   = (float*)(ws + off); off += (size_t)BATCH * 784 * 4;       // 25,690,112 B
    float* qkv  = (float*)(ws + off); off += (size_t)BATCH * 16 * 4;        //    524,288 B
    float* outp = (float*)(ws + off); off += (size_t)BATCH * 4 * 4;         //    131,072 B
    float* stats = (float*)(ws + off);                                      //        32 B

    conv1_kernel<<<(BATCH * 8 * 196 + 255) / 256, 256, 0, stream>>>(x, c1w, c1b, f1);
    conv2_wmma_kernel<<<BATCH, 128, 0, stream>>>(f1, c2w, c2b, f2);
    qkv_wmma_kernel<<<BATCH / 64, 128, 0, stream>>>(f2, qw, qb, qkv);
    attn_out_kernel<<<(BATCH + 255) / 256, 256, 0, stream>>>(params, qkv, ow, ob, outp);
    bn_stats_kernel<<<1, 1024, 0, stream>>>(outp, stats);
    bn_apply_kernel<<<(BATCH * 4 + 255) / 256, 256, 0, stream>>>(outp, stats, gamma, beta,
                                                                 (float*)d_out);
}